// HierarchicalEntropyComputer_82480551953012
// MI455X (gfx1250) — compile-verified
//
#include <hip/hip_runtime.h>
#include <math.h>

typedef float v2f __attribute__((ext_vector_type(2)));
typedef float v8f __attribute__((ext_vector_type(8)));

#define DIM 1024
#define NB  64   // batch
// ws layout (floats): Xw[64*1024] | mean[1024] | logd[2*1024] | Sig[1024^2] | SigR[1024^2]
#define OFF_XW   0
#define OFF_MEAN (OFF_XW + NB * DIM)
#define OFF_LOGD (OFF_MEAN + DIM)
#define OFF_SIG  (OFF_LOGD + 2 * DIM)
#define OFF_SIGR (OFF_SIG + DIM * DIM)

// ---------------------------------------------------------------------------
// 1) pack sites (8,64,128) -> X (64,1024) row-major:  X[b][s*128+c] = sites[s][b][c]
// ---------------------------------------------------------------------------
__global__ void pack_kernel(const float* __restrict__ sites, float* __restrict__ Xw) {
  int idx = blockIdx.x * blockDim.x + threadIdx.x;       // 0 .. 65535
  int b = idx >> 10;
  int j = idx & 1023;
  int s = j >> 7;
  int c = j & 127;
  Xw[idx] = sites[s * (NB * 128) + b * 128 + c];
}

// ---------------------------------------------------------------------------
// 2) column means over the 64 rows
// ---------------------------------------------------------------------------
__global__ void colmean_kernel(const float* __restrict__ Xw, float* __restrict__ mean) {
  int c = blockIdx.x * blockDim.x + threadIdx.x;
  if (c < DIM) {
    float s = 0.f;
    for (int r = 0; r < NB; ++r) s += Xw[r * DIM + c];
    mean[c] = s * (1.0f / (float)NB);
  }
}

// ---------------------------------------------------------------------------
// 3) Sig = Xc^T Xc / 64 + 1e-6 I  via V_WMMA_F32_16X16X4_F32 (fp32 matrix path)
//    Also writes index-reversed copy SigR[i][j] = Sig[n-1-i][n-1-j].
//    One wave per 16x16 output tile; K=64 in 16 steps of 4.
// ---------------------------------------------------------------------------
__global__ void cov_wmma_kernel(const float* __restrict__ Xw,
                                const float* __restrict__ mean,
                                float* __restrict__ Sig,
                                float* __restrict__ SigR) {
  const int n = DIM;
  const int wave = (blockIdx.x * blockDim.x + threadIdx.x) >> 5;  // 0..4095
  const int lane = threadIdx.x & 31;
  const int ti = wave >> 6;           // row tile
  const int tj = wave & 63;           // col tile
  const int m  = lane & 15;
  const int koff = (lane >> 4) << 1;  // 0 for lanes 0-15, 2 for lanes 16-31 (ISA 16x4 layout)
  const int gi = ti * 16 + m;
  const int gj = tj * 16 + m;
  const float mi = mean[gi];
  const float mj = mean[gj];

  v8f acc = {0.f, 0.f, 0.f, 0.f, 0.f, 0.f, 0.f, 0.f};
#pragma unroll
  for (int kk = 0; kk < NB; kk += 4) {
    v2f a, bb;
    a.x  = Xw[(kk + koff    ) * n + gi] - mi;   // A[m][koff]   (A = Xc^T)
    a.y  = Xw[(kk + koff + 1) * n + gi] - mi;   // A[m][koff+1]
    bb.x = Xw[(kk + koff    ) * n + gj] - mj;   // B[koff][m]   (B = Xc)
    bb.y = Xw[(kk + koff + 1) * n + gj] - mj;   // B[koff+1][m]
    acc = __builtin_amdgcn_wmma_f32_16x16x4_f32(false, a, false, bb,
                                                (short)0, acc, false, false);
  }

  const int rowoff = (lane < 16) ? 0 : 8;       // ISA 16x16 f32 C/D layout
#pragma unroll
  for (int r = 0; r < 8; ++r) {
    int grow = ti * 16 + rowoff + r;
    float v = acc[r] * (1.0f / (float)NB);
    if (grow == gj) v += 1e-6f;                  // REG on the diagonal
    Sig[grow * n + gj] = v;
    SigR[(n - 1 - grow) * n + (n - 1 - gj)] = v;
  }
}

// ---------------------------------------------------------------------------
// 4) Two in-place Choleskys (block 0: Sig, block 1: SigR), right-looking,
//    pivot column cached in LDS. logd[blk][k] = 2*log(L_kk).
// ---------------------------------------------------------------------------
__global__ void chol_kernel(float* __restrict__ Sig, float* __restrict__ SigR,
                            float* __restrict__ logd) {
  const int n = DIM;
  float* A  = (blockIdx.x == 0) ? Sig : SigR;
  float* ld = logd + blockIdx.x * n;
  __shared__ float col[DIM];
  __shared__ float sdiag;
  const int tid = threadIdx.x;
  const int nt  = blockDim.x;

  for (int k = 0; k < n; ++k) {
    if (tid == 0) {
      float d = sqrtf(fmaxf(A[k * n + k], 1e-30f));
      A[k * n + k] = d;
      ld[k] = 2.0f * logf(d);
      sdiag = d;
    }
    __syncthreads();
    float invd = 1.0f / sdiag;
    for (int i = k + 1 + tid; i < n; i += nt) {
      float v = A[i * n + k] * invd;
      A[i * n + k] = v;
      col[i] = v;
    }
    __syncthreads();
    for (int i = k + 1 + tid; i < n; i += nt) {
      float lik = col[i];
      float* row = A + i * n;
      for (int j = k + 1; j <= i; ++j) row[j] = fmaf(-lik, col[j], row[j]);
    }
    __syncthreads();
  }
}

// ---------------------------------------------------------------------------
// small symmetric eigen solvers (device, single thread)
// ---------------------------------------------------------------------------
__device__ inline void eig2(float a, float b, float c, float* lam) {
  float half = 0.5f * (a + c);
  float disc = sqrtf(fmaxf(0.25f * (a - c) * (a - c) + b * b, 0.0f));
  lam[0] = half + disc;
  lam[1] = half - disc;
}

__device__ inline void eig3(float a00, float a01, float a02,
                            float a11, float a12, float a22, float* lam) {
  float p1 = a01 * a01 + a02 * a02 + a12 * a12;
  float q = (a00 + a11 + a22) * (1.0f / 3.0f);
  if (p1 < 1e-24f) { lam[0] = a00; lam[1] = a11; lam[2] = a22; return; }
  float p2 = (a00 - q) * (a00 - q) + (a11 - q) * (a11 - q) + (a22 - q) * (a22 - q) + 2.0f * p1;
  float p = sqrtf(p2 * (1.0f / 6.0f));
  float ip = 1.0f / p;
  float b00 = (a00 - q) * ip, b11 = (a11 - q) * ip, b22 = (a22 - q) * ip;
  float b01 = a01 * ip, b02 = a02 * ip, b12 = a12 * ip;
  float detB = b00 * (b11 * b22 - b12 * b12)
             - b01 * (b01 * b22 - b12 * b02)
             + b02 * (b01 * b12 - b11 * b02);
  float r = fminf(fmaxf(0.5f * detB, -1.0f), 1.0f);
  float phi = acosf(r) * (1.0f / 3.0f);
  float e1 = q + 2.0f * p * cosf(phi);
  float e3 = q + 2.0f * p * cosf(phi + 2.0943951023931953f);
  float e2 = 3.0f * q - e1 - e3;
  lam[0] = e1; lam[1] = e2; lam[2] = e3;
}

__device__ inline void eig4(float M[4][4], float* lam) {
  for (int sweep = 0; sweep < 12; ++sweep) {
    for (int p = 0; p < 3; ++p) {
      for (int q = p + 1; q < 4; ++q) {
        float apq = M[p][q];
        if (fabsf(apq) < 1e-14f) continue;
        float theta = 0.5f * (M[q][q] - M[p][p]) / apq;
        float t = 1.0f / (fabsf(theta) + sqrtf(theta * theta + 1.0f));
        if (theta < 0.0f) t = -t;
        float c = 1.0f / sqrtf(t * t + 1.0f);
        float s = t * c;
        for (int k = 0; k < 4; ++k) {
          float akp = M[k][p], akq = M[k][q];
          M[k][p] = c * akp - s * akq;
          M[k][q] = s * akp + c * akq;
        }
        for (int k = 0; k < 4; ++k) {
          float apk = M[p][k], aqk = M[q][k];
          M[p][k] = c * apk - s * aqk;
          M[q][k] = s * apk + c * aqk;
        }
      }
    }
  }
  for (int i = 0; i < 4; ++i) lam[i] = M[i][i];
}

__device__ inline float entropy_eigs(const float* lam, int m, float tr, int dA) {
  float inv = 1.0f / (tr + 1e-10f);
  float ent = 0.f;
  for (int i = 0; i < m; ++i) {
    float wc = fmaxf(lam[i] * inv, 1e-10f);
    ent -= wc * logf(wc);
  }
  // (dA - m) exact-zero eigenvalues of rho clamp to 1e-10: each adds -e*ln(e)
  ent += (float)(dA - m) * 2.3025850929940457e-9f;
  return ent;
}

// ---------------------------------------------------------------------------
// 5) per-batch: Gram matrices (rank trick), tiny eigs, entropies; plus
//    masked prefix sums over log-diagonals for the phi terms; combine.
// ---------------------------------------------------------------------------
__global__ void __launch_bounds__(128)
vn_combine_kernel(const float* __restrict__ Xw, const float* __restrict__ logd,
                  float* __restrict__ out) {
  const int n = DIM;
  const int NT = 128;
  const int b = blockIdx.x;
  const int tid = threadIdx.x;

  __shared__ float psi[DIM];
  __shared__ float red[NT];
  __shared__ float part[19][NT];
  __shared__ float gsum[19];
  __shared__ float psum[11];

  // load row, norm
  float ss = 0.f;
  for (int i = tid; i < n; i += NT) { float v = Xw[b * n + i]; psi[i] = v; ss += v * v; }
  red[tid] = ss;
  __syncthreads();
  for (int s = NT / 2; s > 0; s >>= 1) {
    if (tid < s) red[tid] += red[tid + s];
    __syncthreads();
  }
  float inv = 1.0f / fmaxf(sqrtf(red[0]), 1e-12f);
  __syncthreads();
  for (int i = tid; i < n; i += NT) psi[i] *= inv;
  __syncthreads();

  float acc[10];
  // G4 (dA=256, dB=4, nt=1024): 10 unique entries
  for (int t = 0; t < 10; ++t) acc[t] = 0.f;
  for (int a = tid; a < 256; a += NT) {
    const float* p = psi + 4 * a;
    acc[0] += p[0] * p[0]; acc[1] += p[0] * p[1]; acc[2] += p[0] * p[2]; acc[3] += p[0] * p[3];
    acc[4] += p[1] * p[1]; acc[5] += p[1] * p[2]; acc[6] += p[1] * p[3];
    acc[7] += p[2] * p[2]; acc[8] += p[2] * p[3]; acc[9] += p[3] * p[3];
  }
  for (int t = 0; t < 10; ++t) part[t][tid] = acc[t];
  // G3 (dA=337, dB=3, nt=1011)
  for (int t = 0; t < 6; ++t) acc[t] = 0.f;
  for (int a = tid; a < 337; a += NT) {
    const float* p = psi + 3 * a;
    acc[0] += p[0] * p[0]; acc[1] += p[0] * p[1]; acc[2] += p[0] * p[2];
    acc[3] += p[1] * p[1]; acc[4] += p[1] * p[2]; acc[5] += p[2] * p[2];
  }
  for (int t = 0; t < 6; ++t) part[10 + t][tid] = acc[t];
  // G2 (dA=512, dB=2, nt=1024)
  for (int t = 0; t < 3; ++t) acc[t] = 0.f;
  for (int a = tid; a < 512; a += NT) {
    const float* p = psi + 2 * a;
    acc[0] += p[0] * p[0]; acc[1] += p[0] * p[1]; acc[2] += p[1] * p[1];
  }
  for (int t = 0; t < 3; ++t) part[16 + t][tid] = acc[t];
  __syncthreads();
  if (tid < 19) {
    float s = 0.f;
    for (int t = 0; t < NT; ++t) s += part[tid][t];
    gsum[tid] = s;
  }
  __syncthreads();

  // masked partial sums of log-diagonals for phi (logd0 prefixes p & full; logd1 prefixes n-p)
  const int PA[6] = {256, 337, 512, 686, 768, 1024};
  const int PB[5] = {768, 687, 512, 338, 256};
  float pa[6], pb[5];
  for (int t = 0; t < 6; ++t) pa[t] = 0.f;
  for (int t = 0; t < 5; ++t) pb[t] = 0.f;
  for (int i = tid; i < n; i += NT) {
    float v0 = logd[i];
    float v1 = logd[n + i];
    for (int t = 0; t < 6; ++t) if (i < PA[t]) pa[t] += v0;
    for (int t = 0; t < 5; ++t) if (i < PB[t]) pb[t] += v1;
  }
  for (int t = 0; t < 6; ++t) part[t][tid] = pa[t];
  for (int t = 0; t < 5; ++t) part[6 + t][tid] = pb[t];
  __syncthreads();
  if (tid < 11) {
    float s = 0.f;
    for (int t = 0; t < NT; ++t) s += part[tid][t];
    psum[tid] = s;
  }
  __syncthreads();

  if (tid == 0) {
    float ent0, ent1, ent2;
    { // 4x4 Jacobi
      float M[4][4];
      M[0][0] = gsum[0]; M[0][1] = M[1][0] = gsum[1]; M[0][2] = M[2][0] = gsum[2]; M[0][3] = M[3][0] = gsum[3];
      M[1][1] = gsum[4]; M[1][2] = M[2][1] = gsum[5]; M[1][3] = M[3][1] = gsum[6];
      M[2][2] = gsum[7]; M[2][3] = M[3][2] = gsum[8]; M[3][3] = gsum[9];
      float tr = M[0][0] + M[1][1] + M[2][2] + M[3][3];
      float lam[4];
      eig4(M, lam);
      ent0 = entropy_eigs(lam, 4, tr, 256);
    }
    { // 3x3 closed form
      float tr = gsum[10] + gsum[13] + gsum[15];
      float lam[3];
      eig3(gsum[10], gsum[11], gsum[12], gsum[13], gsum[14], gsum[15], lam);
      ent1 = entropy_eigs(lam, 3, tr, 337);
    }
    { // 2x2 closed form
      float tr = gsum[16] + gsum[18];
      float lam[2];
      eig2(gsum[16], gsum[17], gsum[18], lam);
      ent2 = entropy_eigs(lam, 2, tr, 512);
    }
    float SvN = (ent0 + ent1 + ent2) * (1.0f / 3.0f);
    float Snorm = SvN / (6.2402758451707687f + 1e-8f);   // log(1024/2 + 1)

    float ldfull = psum[5];
    float phimin = 3.4e38f;
    for (int t = 0; t < 5; ++t) {
      float phi = fmaxf(0.5f * (psum[t] + psum[6 + t] - ldfull), 0.0f);
      phimin = fminf(phimin, phi);
    }
    float phinorm = fminf(fmaxf(phimin, 0.0f), 10.0f) * 0.1f;
    out[b] = 0.5f * (Snorm + phinorm);
  }
}

// ---------------------------------------------------------------------------
extern "C" void kernel_launch(void* const* d_in, const int* in_sizes, int n_in,
                              void* d_out, int out_size, void* d_ws, size_t ws_size,
                              hipStream_t stream) {
  (void)in_sizes; (void)n_in; (void)out_size; (void)ws_size;
  const float* sites = (const float*)d_in[0];   // (8,64,128) fp32
  float* out = (float*)d_out;                   // (64,) fp32
  float* ws = (float*)d_ws;

  float* Xw   = ws + OFF_XW;
  float* mean = ws + OFF_MEAN;
  float* logd = ws + OFF_LOGD;
  float* Sig  = ws + OFF_SIG;
  float* SigR = ws + OFF_SIGR;

  pack_kernel<<<(NB * DIM) / 256, 256, 0, stream>>>(sites, Xw);
  colmean_kernel<<<DIM / 256, 256, 0, stream>>>(Xw, mean);
  // 64x64 tiles, one wave each -> 4096 waves; 8 waves/block -> 512 blocks
  cov_wmma_kernel<<<512, 256, 0, stream>>>(Xw, mean, Sig, SigR);
  chol_kernel<<<2, 1024, 0, stream>>>(Sig, SigR, logd);
  vn_combine_kernel<<<NB, 128, 0, stream>>>(Xw, logd, out);
}